// OneHotPooling_34857954574530
// MI455X (gfx1250) — compile-verified
//
#include <hip/hip_runtime.h>
#include <stdint.h>

#define FDIM 128  // filters per event (reference F)

// ---------------------------------------------------------------------------
// helpers
// ---------------------------------------------------------------------------
__device__ __forceinline__ float softplus_f(float x) {
  // numerically stable softplus: max(x,0) + log1p(exp(-|x|))
  return fmaxf(x, 0.0f) + log1pf(__expf(-fabsf(x)));
}

// LDS byte offset of a pointer into a __shared__ object (addrspace(3) offset)
__device__ __forceinline__ unsigned lds_off_u32(void* p) {
  return (unsigned)(unsigned long long)((__attribute__((address_space(3))) char*)p);
}

// native f32 atomic add, no return value used -> non-returning form
__device__ __forceinline__ void atomic_add_f32(float* p, float v) {
  __hip_atomic_fetch_add(p, v, __ATOMIC_RELAXED, __HIP_MEMORY_SCOPE_AGENT);
}

// ---------------------------------------------------------------------------
// pass 1: per-segment event counts (integer atomics, 2 MB of reads)
// ---------------------------------------------------------------------------
__global__ void count_segments(const int* __restrict__ seg,
                               int* __restrict__ counts, int n) {
  int i = blockIdx.x * blockDim.x + threadIdx.x;
  if (i < n) atomicAdd(&counts[seg[i]], 1);
}

// pass 2: reciprocal of max(count,1) so the main pass can pre-scale weights
__global__ void inv_counts_kernel(const int* __restrict__ counts,
                                  float* __restrict__ inv, int n) {
  int i = blockIdx.x * blockDim.x + threadIdx.x;
  if (i < n) inv[i] = 1.0f / fmaxf((float)counts[i], 1.0f);
}

// ---------------------------------------------------------------------------
// pass 3: main gather/exp/scatter. One wave (32 lanes) per event, 4 filters
// per lane. The 256 MB id stream is staged through LDS with double-buffered
// CDNA5 async copies (global_load_async_to_lds_b128 / s_wait_asynccnt), with
// a non-temporal hint so the once-read stream does not evict the L2-resident
// output accumulators (134 MB out + 1 MB gather table fit in 192 MB L2).
// ---------------------------------------------------------------------------
__global__ __launch_bounds__(256) void onehot_pool_main(
    const float* __restrict__ times_in,
    const float* __restrict__ times_out,
    const float* __restrict__ raw,
    const int* __restrict__ seg,
    const int* __restrict__ ids,
    const float* __restrict__ invc,
    float* __restrict__ out,
    int n_full, int rem) {
  __shared__ __align__(16) int lds_ids[2][8 * FDIM / 4];  // 2 x 1 KB

  const int t = threadIdx.x;
  const int lane_f = t & 31;   // lane within wave
  const int e_loc  = t >> 5;   // 0..7: which event of the 8-event chunk
  const int f0     = lane_f * 4;

  // per-thread decay rates for its 4 filters (raw is 512 B, L2-hot)
  const float dA = softplus_f(raw[f0 + 0]);
  const float dB = softplus_f(raw[f0 + 1]);
  const float dC = softplus_f(raw[f0 + 2]);
  const float dD = softplus_f(raw[f0 + 3]);

  auto process4 = [&](int e, const int4 id4) {
    // wave-uniform addresses -> single request per wave
    const float tin = times_in[e];
    const int   s   = seg[e];
    const float ic  = invc[s];
    // weights = exp(-rate*(t_out - t_in)) * 1/max(count,1)
    const float w0 = __expf((tin - times_out[id4.x]) * dA) * ic;
    const float w1 = __expf((tin - times_out[id4.y]) * dB) * ic;
    const float w2 = __expf((tin - times_out[id4.z]) * dC) * ic;
    const float w3 = __expf((tin - times_out[id4.w]) * dD) * ic;
    float* o = out + (size_t)s * FDIM + f0;
    atomic_add_f32(o + 0, w0);
    atomic_add_f32(o + 1, w1);
    atomic_add_f32(o + 2, w2);
    atomic_add_f32(o + 3, w3);
  };

  auto issue_async = [&](int chunk, int buf) {
    // thread t copies bytes [t*16, t*16+16) of the 4 KB chunk
    unsigned loff = lds_off_u32(&lds_ids[buf][t * 4]);
    const int* g = ids + (size_t)chunk * (8 * FDIM) + t * 4;
    // s_wait_dscnt 0: guarantee prior ds_load_b128 of this buffer drained
    // before the async engine overwrites it (WAR across iterations).
    // th:TH_LOAD_NT: stream is single-use; keep it out of L2.
    asm volatile("s_wait_dscnt 0\n\t"
                 "global_load_async_to_lds_b128 %0, %1, off th:TH_LOAD_NT"
                 :: "v"(loff), "v"(g)
                 : "memory");
  };

  int chunk = blockIdx.x;
  if (chunk < n_full) {
    issue_async(chunk, 0);
    int buf = 0;
    for (;;) {
      const int next = chunk + (int)gridDim.x;
      const bool have_next = next < n_full;
      if (have_next) {
        issue_async(next, buf ^ 1);
        // async loads complete in order: cnt<=1 means current tile landed
        asm volatile("s_wait_asynccnt 1" ::: "memory");
      } else {
        asm volatile("s_wait_asynccnt 0" ::: "memory");
      }
      const int4 id4 = *(const int4*)&lds_ids[buf][t * 4];  // ds_load_b128
      process4(chunk * 8 + e_loc, id4);
      if (!have_next) break;
      chunk = next;
      buf ^= 1;
    }
  }

  // generic tail (E_in not divisible by 8): direct global b128 loads
  if (blockIdx.x == 0 && e_loc < rem) {
    const int e = n_full * 8 + e_loc;
    const int4 id4 = *(const int4*)(ids + (size_t)e * FDIM + f0);
    process4(e, id4);
  }
}

// ---------------------------------------------------------------------------
// host launcher
// ---------------------------------------------------------------------------
extern "C" void kernel_launch(void* const* d_in, const int* in_sizes, int n_in,
                              void* d_out, int out_size, void* d_ws, size_t ws_size,
                              hipStream_t stream) {
  const float* times_in  = (const float*)d_in[0];
  const float* times_out = (const float*)d_in[1];
  const float* raw       = (const float*)d_in[2];
  const int*   seg       = (const int*)d_in[3];
  const int*   ids       = (const int*)d_in[4];
  float* out = (float*)d_out;

  const int E_in  = in_sizes[0];
  const int E_out = in_sizes[1];

  int*   counts = (int*)d_ws;
  float* inv    = (float*)((char*)d_ws + (size_t)E_out * sizeof(int));

  // zero accumulators (graph-capture safe)
  hipMemsetAsync(out, 0, (size_t)out_size * sizeof(float), stream);
  hipMemsetAsync(counts, 0, (size_t)E_out * sizeof(int), stream);

  count_segments<<<(E_in + 255) / 256, 256, 0, stream>>>(seg, counts, E_in);
  inv_counts_kernel<<<(E_out + 255) / 256, 256, 0, stream>>>(counts, inv, E_out);

  const int n_full = E_in >> 3;   // 8 events per 4 KB chunk
  const int rem    = E_in & 7;
  int blocks = n_full < 4096 ? (n_full > 0 ? n_full : 1) : 4096;
  onehot_pool_main<<<blocks, 256, 0, stream>>>(times_in, times_out, raw, seg,
                                               ids, inv, out, n_full, rem);
}